// NodeProcessor_1159641170086
// MI455X (gfx1250) — compile-verified
//
#include <hip/hip_runtime.h>
#include <hip/hip_bf16.h>
#include <math.h>

// ---------------------------------------------------------------------------
// NodeProcessor for MI455X (gfx1250):
//   agg = segment_sum(edge_attr, dst)                  (atomic scatter-add)
//   h   = silu(concat(x,agg) @ W1 + b1)                (fp32 WMMA 16x16x4)
//   h   = h @ W2 + b2                                  (fp32 WMMA 16x16x4)
//   out = LayerNorm(h)*gamma + beta + x                (fused, LDS reduce)
// Weights are repacked into k-pair-interleaved layout so every WMMA B operand
// is a single global_load_b64 (instead of 2x global_load_b32 from two rows).
// ---------------------------------------------------------------------------

#define ND     256      // node/edge feature dim
#define NH     512      // hidden dim (= 2*ND)
#define LN_EPS 1e-5f

typedef float v2f __attribute__((ext_vector_type(2)));
typedef float v8f __attribute__((ext_vector_type(8)));

__device__ __forceinline__ v8f wmma_f32(v2f a, v2f b, v8f c) {
  // V_WMMA_F32_16X16X4_F32 : D = A(16x4) * B(4x16) + C(16x16), all fp32
  return __builtin_amdgcn_wmma_f32_16x16x4_f32(
      /*neg_a=*/false, a, /*neg_b=*/false, b,
      /*c_mod=*/(short)0, c, /*reuse_a=*/false, /*reuse_b=*/false);
}

// Accumulate a 16(M) x 64(N) tile; c[t] covers columns nBase + t*16 .. +15.
// act: row-major [*, LDA].  Wp: k-pair packed, pair p at Wp[(p*LDW + n)*2].
// Lane ℓ: m-row = mBase + (ℓ&15); K pair = kk/2 + (ℓ>>4); both A and B read
// the same {k, k+1} pair, so A is one b64 and each B tile is one b64.
template <int LDA, int LDW>
__device__ __forceinline__ void mma_strip(const float* __restrict__ act,
                                          const float* __restrict__ Wp,
                                          int mBase, int nBase, int lane,
                                          int Klen, v8f c[4]) {
  const int kh   = lane >> 4;          // which K half-pair this half-wave owns
  const int mrow = mBase + (lane & 15);
  const int ncol = nBase + (lane & 15);
  #pragma unroll 4
  for (int kk = 0; kk < Klen; kk += 4) {
    const int k = kk + 2 * kh;                                   // even
    const v2f a = *(const v2f*)(act + (size_t)mrow * LDA + k);   // b64
    const float* wp = Wp + ((size_t)(k >> 1) * LDW + ncol) * 2;
    #pragma unroll
    for (int t = 0; t < 4; ++t) {
      const v2f b = *(const v2f*)(wp + t * 32);                  // b64
      c[t] = wmma_f32(a, b, c[t]);
    }
  }
}

// ---- kernel 0: repack W[K][Nc] into k-pair interleaved Wp ----------------
__global__ void k_pack_pairs(const float* __restrict__ W, float* __restrict__ Wp,
                             int Nc, int totalPairs) {
  const int gid = blockIdx.x * blockDim.x + threadIdx.x;   // over (K/2)*Nc
  if (gid >= totalPairs) return;
  const int p = gid / Nc;
  const int n = gid - p * Nc;
  v2f v;
  v.x = W[(size_t)(2 * p) * Nc + n];
  v.y = W[(size_t)(2 * p + 1) * Nc + n];
  *(v2f*)(Wp + (size_t)gid * 2) = v;
}

// ---- kernel 1: zero the aggregation buffer -------------------------------
__global__ void k_zero(float* __restrict__ p, int n) {
  int i = blockIdx.x * blockDim.x + threadIdx.x;
  if (i < n) p[i] = 0.0f;
}

// ---- kernel 2: scatter-add edge features onto destination nodes ----------
// one 256-thread block per edge; coalesced load + coalesced atomic stream
__global__ void k_scatter(const float* __restrict__ edge_attr,
                          const int* __restrict__ dst,   // edge_index row 0
                          float* __restrict__ agg) {
  const int e = blockIdx.x;
  const int c = threadIdx.x;                 // 0..255
  const int d = dst[e];
  atomicAdd(&agg[(size_t)d * ND + c], edge_attr[(size_t)e * ND + c]);
}

// ---- kernel 3: h1 = silu([x|agg] @ W1 + b1) ------------------------------
// 8 waves / block; wave owns one 16-row strip and a 64-col slab (grid.y of 8)
__global__ void k_gemm1_silu(const float* __restrict__ x,
                             const float* __restrict__ agg,
                             const float* __restrict__ W1p,  // packed [512,512]
                             const float* __restrict__ b1,   // [512]
                             float* __restrict__ h1,         // [N,512]
                             int nStrips) {
  const int wave  = threadIdx.x >> 5;
  const int lane  = threadIdx.x & 31;
  const int strip = blockIdx.x * 8 + wave;
  if (strip >= nStrips) return;              // wave-uniform exit, EXEC stays full
  const int mBase = strip * 16;
  const int nBase = blockIdx.y * 64;

  v8f c[4] = {v8f{}, v8f{}, v8f{}, v8f{}};
  // K = 0..255   : x part, W1 rows 0..255   (pairs 0..127)
  mma_strip<ND, NH>(x,   W1p,                   mBase, nBase, lane, ND, c);
  // K = 256..511 : agg part, W1 rows 256..511 (pairs 128..255 -> +ND*NH floats)
  mma_strip<ND, NH>(agg, W1p + (size_t)ND * NH, mBase, nBase, lane, ND, c);

  const int kh = lane >> 4;
  #pragma unroll
  for (int t = 0; t < 4; ++t) {
    const int col  = nBase + t * 16 + (lane & 15);
    const float bs = b1[col];
    #pragma unroll
    for (int r = 0; r < 8; ++r) {
      const int row = mBase + r + 8 * kh;
      float v = c[t][r] + bs;
      v = v * (1.0f / (1.0f + __expf(-v)));          // SiLU
      h1[(size_t)row * NH + col] = v;
    }
  }
}

// ---- kernel 4: out = LN(h1 @ W2 + b2)*gamma + beta + x -------------------
// 4 waves / block cover all 256 output cols of one 16-row strip; LDS reduce.
__global__ void k_gemm2_ln(const float* __restrict__ h1,    // [N,512]
                           const float* __restrict__ W2p,   // packed [512,256]
                           const float* __restrict__ b2,    // [256]
                           const float* __restrict__ gamma,
                           const float* __restrict__ beta,
                           const float* __restrict__ x,     // [N,256]
                           float* __restrict__ out) {
  __shared__ float sh[16 * ND];          // 16 rows x 256 cols of h2
  __shared__ float red[2 * 16 * 8];      // per-row partial sums / sumsq

  const int wave  = threadIdx.x >> 5;    // 0..3
  const int lane  = threadIdx.x & 31;
  const int strip = blockIdx.x;
  const int mBase = strip * 16;
  const int nBase = wave * 64;

  v8f c[4] = {v8f{}, v8f{}, v8f{}, v8f{}};
  mma_strip<NH, ND>(h1, W2p, mBase, nBase, lane, NH, c);

  const int kh = lane >> 4;
  #pragma unroll
  for (int t = 0; t < 4; ++t) {
    const int col  = nBase + t * 16 + (lane & 15);
    const float bs = b2[col];
    #pragma unroll
    for (int r = 0; r < 8; ++r) {
      sh[(r + 8 * kh) * ND + col] = c[t][r] + bs;
    }
  }
  __syncthreads();

  // LayerNorm: 8 threads per row, 32 cols each
  const int row = threadIdx.x >> 3;      // 0..15
  const int seg = threadIdx.x & 7;       // 0..7
  float s = 0.0f, s2 = 0.0f;
  #pragma unroll 8
  for (int j = 0; j < 32; ++j) {
    const float v = sh[row * ND + seg * 32 + j];
    s += v; s2 += v * v;
  }
  red[row * 8 + seg]       = s;
  red[128 + row * 8 + seg] = s2;
  __syncthreads();

  float mean = 0.0f, msq = 0.0f;
  #pragma unroll
  for (int p = 0; p < 8; ++p) {
    mean += red[row * 8 + p];
    msq  += red[128 + row * 8 + p];
  }
  mean *= (1.0f / ND);
  const float var = msq * (1.0f / ND) - mean * mean;
  const float inv = rsqrtf(var + LN_EPS);

  const int grow = mBase + row;
  #pragma unroll 8
  for (int j = 0; j < 32; ++j) {
    const int col = seg * 32 + j;
    const float v = sh[row * ND + col];
    out[(size_t)grow * ND + col] =
        (v - mean) * inv * gamma[col] + beta[col] + x[(size_t)grow * ND + col];
  }
}

// ---------------------------------------------------------------------------
extern "C" void kernel_launch(void* const* d_in, const int* in_sizes, int n_in,
                              void* d_out, int out_size, void* d_ws, size_t ws_size,
                              hipStream_t stream) {
  const float* x     = (const float*)d_in[0];   // [N,256]
  const int*   eidx  = (const int*)  d_in[1];   // [2,E], row0 = dst
  const float* eattr = (const float*)d_in[2];   // [E,256]
  const float* W1    = (const float*)d_in[3];   // [512,512]
  const float* b1    = (const float*)d_in[4];   // [512]
  const float* W2    = (const float*)d_in[5];   // [512,256]
  const float* b2    = (const float*)d_in[6];   // [256]
  const float* gamma = (const float*)d_in[7];   // [256]
  const float* beta  = (const float*)d_in[8];   // [256]
  float*       out   = (float*)d_out;

  const int Nn = in_sizes[0] / ND;              // 50000
  const int E  = in_sizes[2] / ND;              // 800000
  const int nStrips = Nn / 16;                  // 3125 (50000 divides exactly)

  float* agg = (float*)d_ws;                    // [N,256]   51.2 MB
  float* h1  = agg + (size_t)Nn * ND;           // [N,512]  102.4 MB
  float* W1p = h1  + (size_t)Nn * NH;           // 512*512   1.0 MB
  float* W2p = W1p + (size_t)NH * NH;           // 512*256   0.5 MB

  // 0) repack weights into k-pair interleaved layout (B operand -> one b64)
  {
    const int p1 = (NH / 2) * NH;               // 131072 pairs
    const int p2 = (NH / 2) * ND;               //  65536 pairs
    k_pack_pairs<<<(p1 + 255) / 256, 256, 0, stream>>>(W1, W1p, NH, p1);
    k_pack_pairs<<<(p2 + 255) / 256, 256, 0, stream>>>(W2, W2p, ND, p2);
  }
  // 1) agg = 0
  {
    const int n = Nn * ND;
    k_zero<<<(n + 255) / 256, 256, 0, stream>>>(agg, n);
  }
  // 2) scatter-add edges
  k_scatter<<<E, ND, 0, stream>>>(eattr, eidx, agg);
  // 3) GEMM1 + SiLU  (grid.y: 512 cols / 64)
  {
    dim3 grid((nStrips + 7) / 8, NH / 64);
    k_gemm1_silu<<<grid, 256, 0, stream>>>(x, agg, W1p, b1, h1, nStrips);
  }
  // 4) GEMM2 + bias + LayerNorm + residual
  k_gemm2_ln<<<nStrips, 128, 0, stream>>>(h1, W2p, b2, gamma, beta, x, out);
}